// RetentionMixerModule_80032420593807
// MI455X (gfx1250) — compile-verified
//
#include <hip/hip_runtime.h>

// Problem constants (fixed by reference: x[8,4096,2048] fp32, decay[2048])
#define B_N 8
#define T_N 4096
#define D_N 2048
#define N_CHUNK 16
#define CHUNK_LEN (T_N / N_CHUNK)   // 256
#define BLK_D 256                   // threads per block == channels per block
#define TILE_T 16                   // time rows staged per LDS tile
#define N_TILE (CHUNK_LEN / TILE_T) // 16 tiles per chunk
#define XFERS (TILE_T * BLK_D / 4 / BLK_D) // 4 x b128 issues per thread per tile

// Exact pointee type the builtin expects: int __attribute__((vector_size(16)))
typedef int v4i __attribute__((vector_size(16)));
typedef __attribute__((address_space(1))) v4i* gv4i_ptr;  // global (prints as __device__)
typedef __attribute__((address_space(3))) v4i* lv4i_ptr;  // LDS

// generic LDS pointer's low 32 bits == LDS byte offset (ISA: LDS aperture addr[31:0])
#define AS1_V4I(p) ((gv4i_ptr)(unsigned long long)(p))
#define AS3_V4I(p) ((lv4i_ptr)(unsigned int)(unsigned long long)(p))

__device__ __forceinline__ void async_copy_b128(const float* g, float* l) {
#if __has_builtin(__builtin_amdgcn_global_load_async_to_lds_b128)
  __builtin_amdgcn_global_load_async_to_lds_b128(AS1_V4I(g), AS3_V4I(l), 0, 0);
#else
  asm volatile("global_load_async_to_lds_b128 %0, %1, off"
               :: "v"((unsigned int)(unsigned long long)(l)),
                  "v"((unsigned long long)(g))
               : "memory");
#endif
}

template <int N>
__device__ __forceinline__ void wait_async() {
#if __has_builtin(__builtin_amdgcn_s_wait_asynccnt)
  __builtin_amdgcn_s_wait_asynccnt(N);
#else
  asm volatile("s_wait_asynccnt %0" :: "n"(N) : "memory");
#endif
}

__device__ __forceinline__ float sigmoid_f(float p) {
  return 1.0f / (1.0f + __expf(-p));
}

// Stage one TILE_T x BLK_D tile of x into an LDS buffer with async b128 copies.
// gbase points at (row0, d0) of the tile; row stride is D_N floats.
__device__ __forceinline__ void stage_tile(const float* __restrict__ gbase,
                                           float* __restrict__ lbuf, int tid) {
#pragma unroll
  for (int i = 0; i < XFERS; ++i) {
    int q  = tid + i * BLK_D;     // 0..1023 float4-chunks in tile
    int r  = q >> 6;              // 64 float4 per row of 256 floats
    int c4 = (q & 63) << 2;       // float column within row
    async_copy_b128(gbase + (size_t)r * D_N + c4, lbuf + r * BLK_D + c4);
  }
}

// Shared chunk-scan body. WRITE_Y=false: zero-init, emit local final to ws.
// WRITE_Y=true: seed from ws carry-in, stream y out.
template <bool WRITE_Y>
__device__ __forceinline__ void scan_chunk_body(
    const float* __restrict__ x, const float* __restrict__ decay_param,
    float* __restrict__ y, float* __restrict__ ws) {
  __shared__ float lds[2][TILE_T * BLK_D];  // 32 KB double-buffered tile

  const int tid  = threadIdx.x;
  const int dblk = blockIdx.x;   // 0..D_N/BLK_D-1
  const int c    = blockIdx.y;   // chunk
  const int b    = blockIdx.z;   // batch
  const int d    = dblk * BLK_D + tid;

  const float dc = sigmoid_f(decay_param[d]);
  const float om = 1.0f - dc;

  const int t0 = c * CHUNK_LEN;
  const float* xbase = x + ((size_t)b * T_N + t0) * D_N + (size_t)dblk * BLK_D;

  // Kick off tile 0 DMA before touching anything else.
  stage_tile(xbase, lds[0], tid);

  const size_t cidx = ((size_t)b * N_CHUNK + c) * D_N + d;
  float h = 0.0f;
  if (WRITE_Y) h = ws[cidx];  // carry-in (exclusive-scanned chunk state)

  float* ybase = WRITE_Y ? (y + ((size_t)b * T_N + t0) * D_N + d) : nullptr;

  for (int k = 0; k < N_TILE; ++k) {
    if (k + 1 < N_TILE) {
      stage_tile(xbase + (size_t)(k + 1) * TILE_T * D_N, lds[(k + 1) & 1], tid);
      wait_async<XFERS>();  // previous tile landed; next batch stays in flight
    } else {
      wait_async<0>();
    }
    __syncthreads();  // publish this wave's async LDS writes to all waves

    const float* cur = lds[k & 1];
#pragma unroll
    for (int r = 0; r < TILE_T; ++r) {
      float xv = cur[r * BLK_D + tid];
      h = dc * xv + om * h;
      if (WRITE_Y) ybase[(size_t)(k * TILE_T + r) * D_N] = h;
    }
    __syncthreads();  // everyone done reading before buffer is re-targeted
  }

  if (!WRITE_Y) ws[cidx] = h;  // local final state of this chunk
}

// Pass 1: per-chunk scan with zero init; writes each chunk's final state to ws.
__global__ __launch_bounds__(BLK_D) void retn_pass1_partials(
    const float* __restrict__ x, const float* __restrict__ decay_param,
    float* __restrict__ ws) {
  scan_chunk_body<false>(x, decay_param, nullptr, ws);
}

// Pass 2: exclusive scan of chunk states along the chunk axis, per (b,d).
// carry_out(chunk c) = local_final(c) + (1-d)^CHUNK_LEN * carry_in(c)
__global__ __launch_bounds__(256) void retn_pass2_carries(
    const float* __restrict__ decay_param, float* __restrict__ ws) {
  const int gid = blockIdx.x * 256 + threadIdx.x;  // 0 .. B_N*D_N-1
  const int b = gid / D_N;
  const int d = gid - b * D_N;

  const float dc = sigmoid_f(decay_param[d]);
  float omL = 1.0f - dc;
#pragma unroll
  for (int i = 0; i < 8; ++i) omL *= omL;  // (1-d)^256 == (1-d)^CHUNK_LEN

  float carry = 0.0f;
  for (int c = 0; c < N_CHUNK; ++c) {
    const size_t idx = ((size_t)b * N_CHUNK + c) * D_N + d;
    const float s = ws[idx];  // local final of chunk c
    ws[idx] = carry;          // becomes carry-in for chunk c
    carry = s + omL * carry;  // state at end of chunk c
  }
}

// Pass 3: per-chunk scan seeded with the carry-in; streams y.
__global__ __launch_bounds__(BLK_D) void retn_pass3_final(
    const float* __restrict__ x, const float* __restrict__ decay_param,
    float* __restrict__ y, float* __restrict__ ws) {
  scan_chunk_body<true>(x, decay_param, y, ws);
}

extern "C" void kernel_launch(void* const* d_in, const int* in_sizes, int n_in,
                              void* d_out, int out_size, void* d_ws, size_t ws_size,
                              hipStream_t stream) {
  (void)in_sizes; (void)n_in; (void)out_size; (void)ws_size;
  const float* x  = (const float*)d_in[0];
  const float* dp = (const float*)d_in[1];
  float* y  = (float*)d_out;
  float* ws = (float*)d_ws;  // needs B_N*N_CHUNK*D_N*4 = 1 MB scratch

  dim3 grid(D_N / BLK_D, N_CHUNK, B_N);  // 8 x 16 x 8 = 1024 blocks of 8 waves
  retn_pass1_partials<<<grid, BLK_D, 0, stream>>>(x, dp, ws);
  retn_pass2_carries<<<(B_N * D_N) / 256, 256, 0, stream>>>(dp, ws);
  retn_pass3_final<<<grid, BLK_D, 0, stream>>>(x, dp, y, ws);
}